// CostVolume_4363686773295
// MI455X (gfx1250) — compile-verified
//
#include <hip/hip_runtime.h>
#include <hip/hip_bf16.h>

// CDNA5 cost-volume: banded fp32 GEMM per (b,h) row using V_WMMA_F32_16X16X4_F32,
// TDM (tensor_load_to_lds) staging with hardware LDS row padding, and an
// LDS-staged coalesced epilogue.
// cost[b,h,j,d] = (1/128) * dot(L[b,h,j,:], R[b,h,j-d,:]) for j>=d else 0.

typedef __attribute__((ext_vector_type(2))) float v2f;
typedef __attribute__((ext_vector_type(8))) float v8f;
typedef unsigned int u32;
typedef __attribute__((ext_vector_type(4))) u32 u32x4;
typedef __attribute__((ext_vector_type(8))) u32 u32x8;

#define NB 8
#define NH 160
#define NW 320
#define NC 128
#define ND 64
#define PITCH 132   // 128 + 4 floats pad: TDM pad_interval=6 (128 DW), pad_amount=3 (4 DW)
#define OPITCH 96   // output scatter region pitch (64 rows x 96 slots, d+16 in [1,95])

__global__ __launch_bounds__(128) void costvol_wmma_f32(
    const float* __restrict__ Lf, const float* __restrict__ Rf,
    float* __restrict__ out) {
  __shared__ float ldsL[64 * PITCH];    // L rows J0 .. J0+63
  __shared__ float ldsR[128 * PITCH];   // R rows J0-64 .. J0+63 (p<0 rows left stale; masked)

  const int row = blockIdx.y;           // b*NH + h
  const int J0  = blockIdx.x * 64;      // j-block start
  const long inBase  = (long)row * (NW * NC);
  const long outBase = (long)row * ((long)NW * ND);
  const int tid  = threadIdx.x;         // 0..127
  const int lane = tid & 31;
  const int wv   = tid >> 5;            // wave id 0..3

  // Low 32 bits of a flat LDS pointer are the LDS byte address (ISA 10.2 aperture rule).
  const unsigned ldsLBase = (unsigned)(uintptr_t)(&ldsL[0]);
  const unsigned ldsRBase = (unsigned)(uintptr_t)(&ldsR[0]);

  // ---- TDM staging: wave 0 issues two descriptor-driven DMAs (TENSORcnt-tracked).
  if (wv == 0) {
    // shared group1 word0: data_size=4B, pad_enable, pad_interval=128 DW, pad_amount=4 DW
    const u32 g1w0 = (2u << 16) | (1u << 20) | (6u << 22) | (3u << 25);

    // L tile: 128 x 64 rows at row J0 (always fully in-bounds)
    const unsigned long gaL = (unsigned long)(uintptr_t)(Lf + inBase + (long)J0 * NC);
    const u32x4 g0L = { 1u, ldsLBase, (u32)gaL,
                       ((u32)(gaL >> 32) & 0x01FFFFFFu) | (2u << 30) };
    const u32x8 g1L = { g1w0,
                        128u << 16,          // tensor_dim0 = 128
                        64u << 16,           // tensor_dim1 = 64 rows
                        128u << 16,          // tile_dim0 = 128
                        64u,                 // tile_dim1 = 64 rows (tile_dim2 = 0)
                        128u,                // tensor_dim0_stride = 128 elements
                        0u, 0u };
    asm volatile("tensor_load_to_lds %0, %1" :: "s"(g0L), "s"(g1L) : "memory");

    // R tile: rows max(J0-64,0) .. J0+63. At the left edge load only the 64 valid
    // rows into the upper half of ldsR; the stale lower rows feed only d>j slots,
    // which the copy-out masks to 0.
    const int p0 = J0 - 64;
    const int rlo = (p0 < 0) ? 0 : p0;
    const u32 nrows = (p0 < 0) ? 64u : 128u;
    const u32 ldsROff = ldsRBase + ((p0 < 0) ? (64u * PITCH * 4u) : 0u);
    const unsigned long gaR = (unsigned long)(uintptr_t)(Rf + inBase + (long)rlo * NC);
    const u32x4 g0R = { 1u, ldsROff, (u32)gaR,
                       ((u32)(gaR >> 32) & 0x01FFFFFFu) | (2u << 30) };
    const u32x8 g1R = { g1w0,
                        128u << 16,          // tensor_dim0 = 128
                        nrows << 16,         // tensor_dim1 = nrows
                        128u << 16,          // tile_dim0 = 128
                        nrows,               // tile_dim1 = nrows
                        128u,                // tensor_dim0_stride = 128 elements
                        0u, 0u };
    asm volatile("tensor_load_to_lds %0, %1" :: "s"(g0R), "s"(g1R) : "memory");

    __builtin_amdgcn_s_wait_tensorcnt(0);   // drain before signaling the barrier
  }
  __syncthreads();

  // ---- compute: wave wv owns j-tile j0 = J0 + 16*wv, 5 p-tiles p0 = j0 - 16t
  const int half = lane >> 4;           // lanes>=16 hold K=2,3 of each K-group
  const int n    = lane & 15;           // M for A, N for B

  const int aoff = (wv * 16 + n) * PITCH + 2 * half;
  const int boff = ((wv + 4) * 16 + n) * PITCH + 2 * half;  // t=0 tile (p0 = j0)

  v8f acc0 = {}, acc1 = {}, acc2 = {}, acc3 = {}, acc4 = {};
#pragma unroll 4
  for (int kb = 0; kb < NC; kb += 4) {
    const v2f a  = *(const v2f*)(&ldsL[aoff + kb]);
    const v2f b0 = *(const v2f*)(&ldsR[boff + kb]);
    const v2f b1 = *(const v2f*)(&ldsR[boff - 16 * PITCH + kb]);
    const v2f b2 = *(const v2f*)(&ldsR[boff - 32 * PITCH + kb]);
    const v2f b3 = *(const v2f*)(&ldsR[boff - 48 * PITCH + kb]);
    const v2f b4 = *(const v2f*)(&ldsR[boff - 64 * PITCH + kb]);
    acc0 = __builtin_amdgcn_wmma_f32_16x16x4_f32(false, a, false, b0, (short)0, acc0, false, false);
    acc1 = __builtin_amdgcn_wmma_f32_16x16x4_f32(false, a, false, b1, (short)0, acc1, false, false);
    acc2 = __builtin_amdgcn_wmma_f32_16x16x4_f32(false, a, false, b2, (short)0, acc2, false, false);
    acc3 = __builtin_amdgcn_wmma_f32_16x16x4_f32(false, a, false, b3, (short)0, acc3, false, false);
    acc4 = __builtin_amdgcn_wmma_f32_16x16x4_f32(false, a, false, b4, (short)0, acc4, false, false);
  }

  // ---- epilogue: diagonal scatter into LDS (each (jr,d) slot written exactly once),
  // then branch-free coalesced b128 copy-out.
  __syncthreads();                      // all ldsR reads done before reuse
  float* so = ldsR;                     // 64 x OPITCH scatter region (24 KB)

  // slot(jr,d) = jr*96 + d + 16; jr = 16*wv + m, m = r + 8*half, d = 16t + m - n
  //  -> index = wv*1536 + half*776 + 16 - n + r*97 + 16t   (immediate-offset stores)
  const int soBase = wv * 1536 + half * 776 + 16 - n;
#pragma unroll
  for (int r = 0; r < 8; ++r) {
    so[soBase + r * 97 +  0] = acc0[r];
    so[soBase + r * 97 + 16] = acc1[r];
    so[soBase + r * 97 + 32] = acc2[r];
    so[soBase + r * 97 + 48] = acc3[r];
    so[soBase + r * 97 + 64] = acc4[r];
  }
  __syncthreads();

  const float sc = 1.0f / (float)NC;
  if (J0 != 0) {                        // j = J0+jr >= 64 > d always: no masking
#pragma unroll
    for (int it = 0; it < 8; ++it) {
      const int idx = tid + 128 * it;
      const int jr = idx >> 4;
      const int c4 = (idx & 15) << 2;   // d base
      const float4 v = *(const float4*)(&so[jr * OPITCH + 16 + c4]);
      float4 w;
      w.x = v.x * sc; w.y = v.y * sc; w.z = v.z * sc; w.w = v.w * sc;
      *(float4*)(out + outBase + (long)(J0 + jr) * ND + c4) = w;
    }
  } else {                              // left edge: zero where j < d (ref masks wrap)
#pragma unroll
    for (int it = 0; it < 8; ++it) {
      const int idx = tid + 128 * it;
      const int jr = idx >> 4;          // == j
      const int c4 = (idx & 15) << 2;
      const float4 v = *(const float4*)(&so[jr * OPITCH + 16 + c4]);
      float4 w;
      w.x = (jr >= c4 + 0) ? v.x * sc : 0.0f;
      w.y = (jr >= c4 + 1) ? v.y * sc : 0.0f;
      w.z = (jr >= c4 + 2) ? v.z * sc : 0.0f;
      w.w = (jr >= c4 + 3) ? v.w * sc : 0.0f;
      *(float4*)(out + outBase + (long)jr * ND + c4) = w;
    }
  }
}

extern "C" void kernel_launch(void* const* d_in, const int* in_sizes, int n_in,
                              void* d_out, int out_size, void* d_ws, size_t ws_size,
                              hipStream_t stream) {
  (void)in_sizes; (void)n_in; (void)out_size; (void)d_ws; (void)ws_size;
  const float* L = (const float*)d_in[0];
  const float* R = (const float*)d_in[1];
  float* out = (float*)d_out;
  dim3 grid(NW / 64, NB * NH);  // 5 j-blocks x 1280 (b,h) rows
  costvol_wmma_f32<<<grid, dim3(128), 0, stream>>>(L, R, out);
}